// GNN_node_16578573763066
// MI455X (gfx1250) — compile-verified
//
#include <hip/hip_runtime.h>

#define Nn 100000
#define Ee 1280000
#define Dd 64
#define Ll 5
#define Gg 512

typedef __attribute__((ext_vector_type(2))) float v2f;
typedef __attribute__((ext_vector_type(8))) float v8f;

// ---------------------------------------------------------------- zero
__global__ void zero_kernel(float* __restrict__ p, int n) {
    int i = blockIdx.x * blockDim.x + threadIdx.x;
    if (i < n) p[i] = 0.f;
}

// ------------------------------------------------------- atom encoder
// h[n,c] = sum_f atom_emb[f, x_feats[n,f], c]
__global__ void atom_encoder_kernel(const int* __restrict__ xf,
                                    const float* __restrict__ aemb,
                                    float* __restrict__ h) {
    int gid = blockIdx.x * blockDim.x + threadIdx.x;
    if (gid >= Nn * Dd) return;
    int n = gid >> 6, c = gid & 63;
    float s = 0.f;
#pragma unroll
    for (int f = 0; f < 9; ++f) {
        int v = xf[n * 9 + f];
        s += aemb[(f * 120 + v) * 64 + c];
    }
    h[gid] = s;
}

// ---------------------------------------------------- z = (1+eps_l)*h
__global__ void zinit_kernel(const float* __restrict__ h,
                             float* __restrict__ z,
                             const float* __restrict__ eps, int l) {
    int i = blockIdx.x * blockDim.x + threadIdx.x;
    if (i >= Nn * Dd / 4) return;
    float s = 1.f + eps[l];
    float4 v = ((const float4*)h)[i];
    v.x *= s; v.y *= s; v.z *= s; v.w *= s;
    ((float4*)z)[i] = v;
}

// ------------------------- bond encoder + message + scatter-add to z
// 16 threads per edge, each owns 4 channels (float4).
__global__ void edge_kernel(const float* __restrict__ h,
                            float* __restrict__ z,
                            const int* __restrict__ eidx,
                            const int* __restrict__ eattr,
                            const float* __restrict__ bemb, int l) {
    int gid = blockIdx.x * blockDim.x + threadIdx.x;
    int e = gid >> 4, q = gid & 15;
    if (e >= Ee) return;
    float4 ev = make_float4(0.f, 0.f, 0.f, 0.f);
#pragma unroll
    for (int f = 0; f < 3; ++f) {
        int v = eattr[e * 3 + f];
        float4 tv = ((const float4*)(bemb + (((l * 3 + f) * 6 + v) * 64)))[q];
        ev.x += tv.x; ev.y += tv.y; ev.z += tv.z; ev.w += tv.w;
    }
    int row = eidx[e];        // source
    int col = eidx[Ee + e];   // target
    float4 hv = ((const float4*)(h + row * 64))[q];
    float m0 = fmaxf(hv.x + ev.x, 0.f);
    float m1 = fmaxf(hv.y + ev.y, 0.f);
    float m2 = fmaxf(hv.z + ev.z, 0.f);
    float m3 = fmaxf(hv.w + ev.w, 0.f);
    float* dst = z + col * 64 + q * 4;
    atomicAdd(dst + 0, m0);
    atomicAdd(dst + 1, m1);
    atomicAdd(dst + 2, m2);
    atomicAdd(dst + 3, m3);
}

// --------------- GEMM (N x 64) @ (64 x 64) + bias, fused BN-stat sums
// Optional pre-transform on input: x = relu(x*scale[c] + shift[c]).
// One wave -> 16 rows x 64 cols via V_WMMA_F32_16X16X4_F32 (4 N-tiles).
// W staged in paired-K layout so each B fragment is one aligned ds_load_b64.
__global__ __launch_bounds__(256) void gemm_bn_kernel(
    const float* __restrict__ X, const float* __restrict__ W,
    const float* __restrict__ bias, const float* __restrict__ pre,
    float* __restrict__ Y, float* __restrict__ stats, int nrows) {
    __shared__ float Wp[32 * 128];     // Wp[k>>1][n*2 + (k&1)] = W[k][n]
    __shared__ float bl[64];
    __shared__ float xs[8][16 * 66];   // pitch 66 -> no bank conflicts on A reads
    __shared__ float ssum[64];
    __shared__ float ssq[64];

    int tid = threadIdx.x;
#pragma unroll
    for (int i = 0; i < 16; ++i) {
        int g = tid + i * 256;
        int k = g >> 6, n = g & 63;
        Wp[(k >> 1) * 128 + n * 2 + (k & 1)] = W[g];
    }
    if (tid < 64) { bl[tid] = bias[tid]; ssum[tid] = 0.f; ssq[tid] = 0.f; }

    int wave = tid >> 5, lane = tid & 31;
    int rowbase = blockIdx.x * 128 + wave * 16;
    int c0 = lane * 2;

    float sc0 = 1.f, sh0 = 0.f, sc1 = 1.f, sh1 = 0.f;
    bool hasPre = (pre != nullptr);
    if (hasPre) {
        sc0 = pre[c0];      sc1 = pre[c0 + 1];
        sh0 = pre[64 + c0]; sh1 = pre[64 + c0 + 1];
    }
    // stage this wave's 16x64 row panel (zero-padded past nrows)
#pragma unroll 4
    for (int r = 0; r < 16; ++r) {
        int row = rowbase + r;
        float vx = 0.f, vy = 0.f;
        if (row < nrows) {
            v2f v = *(const v2f*)(X + row * 64 + c0);
            vx = v[0]; vy = v[1];
            if (hasPre) {
                vx = fmaxf(vx * sc0 + sh0, 0.f);
                vy = fmaxf(vy * sc1 + sh1, 0.f);
            }
        }
        xs[wave][r * 66 + c0]     = vx;
        xs[wave][r * 66 + c0 + 1] = vy;
    }
    __syncthreads();

    int nl = lane & 15;            // M index (A) / N index within tile (B,C)
    int khalf = (lane >> 4) * 2;   // K sub-offset for this half-wave (0 or 2)
    v8f acc[4];
#pragma unroll
    for (int nt = 0; nt < 4; ++nt) {
        float bcol = bl[nt * 16 + nl];
#pragma unroll
        for (int g = 0; g < 8; ++g) acc[nt][g] = bcol;
    }

#pragma unroll
    for (int kb = 0; kb < 16; ++kb) {
        int k0 = kb * 4 + khalf;               // even
        v2f a = *(const v2f*)&xs[wave][nl * 66 + k0];
        int kp = (k0 >> 1) * 128;              // paired-K row base
#pragma unroll
        for (int nt = 0; nt < 4; ++nt) {
            v2f b = *(const v2f*)&Wp[kp + (nt * 16 + nl) * 2];
            acc[nt] = __builtin_amdgcn_wmma_f32_16x16x4_f32(
                false, a, false, b, (short)0, acc[nt], false, false);
        }
    }

    // store + per-column sum / sumsq (for batchnorm), masked to valid rows
    int rhalf = (lane >> 4) * 8;
#pragma unroll
    for (int nt = 0; nt < 4; ++nt) {
        int col = nt * 16 + nl;
        float ps = 0.f, pq = 0.f;
#pragma unroll
        for (int g = 0; g < 8; ++g) {
            int row = rowbase + rhalf + g;
            if (row < nrows) {
                float v = acc[nt][g];
                Y[row * 64 + col] = v;
                ps += v; pq += v * v;
            }
        }
        atomicAdd(&ssum[col], ps);
        atomicAdd(&ssq[col], pq);
    }
    __syncthreads();
    if (tid < 64) {
        atomicAdd(&stats[tid], ssum[tid]);
        atomicAdd(&stats[64 + tid], ssq[tid]);
    }
}

// ------------------------------------- BN finalize: scale/shift per ch
__global__ void finalize_kernel(const float* __restrict__ stats,
                                const float* __restrict__ g,
                                const float* __restrict__ b,
                                float* __restrict__ scsh, float invn) {
    int c = threadIdx.x;
    if (c >= 64) return;
    float m = stats[c] * invn;
    float var = stats[64 + c] * invn - m * m;
    var = fmaxf(var, 0.f);
    float rs = rsqrtf(var + 1e-5f);
    float sc = g[c] * rs;
    scsh[c] = sc;
    scsh[64 + c] = b[c] - m * sc;
}

// ----------------- apply BN2 (+relu), write h, pool into fingerprints
__global__ void bn_pool_kernel(const float* __restrict__ zin,
                               const float* __restrict__ scsh,
                               const int* __restrict__ batch,
                               float* __restrict__ hout,
                               float* __restrict__ fp, int l, int dorelu) {
    int gid = blockIdx.x * blockDim.x + threadIdx.x;
    if (gid >= Nn * Dd) return;
    int n = gid >> 6, c = gid & 63;
    float y = zin[gid] * scsh[c] + scsh[64 + c];
    if (dorelu) y = fmaxf(y, 0.f);
    hout[gid] = y;
    atomicAdd(&fp[(batch[n] * Ll + l) * 64 + c], y);
}

// ---------------------------------------------------------------- host
extern "C" void kernel_launch(void* const* d_in, const int* in_sizes, int n_in,
                              void* d_out, int out_size, void* d_ws, size_t ws_size,
                              hipStream_t stream) {
    const int*   x_feats   = (const int*)d_in[0];
    const int*   edge_idx  = (const int*)d_in[1];
    const int*   edge_attr = (const int*)d_in[2];
    const int*   batch     = (const int*)d_in[3];
    const float* atom_emb  = (const float*)d_in[4];
    const float* bond_emb  = (const float*)d_in[5];
    const float* eps       = (const float*)d_in[6];
    const float* W1        = (const float*)d_in[7];
    const float* b1        = (const float*)d_in[8];
    const float* bn1_g     = (const float*)d_in[9];
    const float* bn1_b     = (const float*)d_in[10];
    const float* W2        = (const float*)d_in[11];
    const float* b2        = (const float*)d_in[12];
    const float* bn_g      = (const float*)d_in[13];
    const float* bn_b      = (const float*)d_in[14];
    float* out = (float*)d_out;

    const int ND = Nn * Dd;
    float* ws    = (float*)d_ws;
    float* h     = ws;
    float* z     = ws + ND;
    float* y1    = ws + 2 * ND;
    float* stats = ws + 3 * ND;        // 256 floats: sum1,sq1,sum2,sq2
    float* scsh  = stats + 256;        // 256 floats: scale1,shift1,scale2,shift2

    float* fp = out + ND;              // [G, L, D] accumulator region
    const float invn = 1.0f / (float)Nn;

    // zero fingerprint region (d_out is poisoned)
    zero_kernel<<<(Gg * Ll * Dd + 255) / 256, 256, 0, stream>>>(fp, Gg * Ll * Dd);

    atom_encoder_kernel<<<(ND + 255) / 256, 256, 0, stream>>>(x_feats, atom_emb, h);

    for (int l = 0; l < Ll; ++l) {
        zero_kernel<<<1, 256, 0, stream>>>(stats, 256);
        zinit_kernel<<<(ND / 4 + 255) / 256, 256, 0, stream>>>(h, z, eps, l);
        edge_kernel<<<(Ee * 16) / 256, 256, 0, stream>>>(h, z, edge_idx, edge_attr,
                                                         bond_emb, l);
        // z @ W1 + b1 -> y1, accumulate BN1 stats
        gemm_bn_kernel<<<(Nn + 127) / 128, 256, 0, stream>>>(
            z, W1 + l * 64 * 64, b1 + l * 64, nullptr, y1, stats, Nn);
        finalize_kernel<<<1, 64, 0, stream>>>(stats, bn1_g + l * 64, bn1_b + l * 64,
                                              scsh, invn);
        // relu(BN1(y1)) @ W2 + b2 -> z, accumulate BN2 stats
        gemm_bn_kernel<<<(Nn + 127) / 128, 256, 0, stream>>>(
            y1, W2 + l * 64 * 64, b2 + l * 64, scsh, z, stats + 128, Nn);
        finalize_kernel<<<1, 64, 0, stream>>>(stats + 128, bn_g + l * 64,
                                              bn_b + l * 64, scsh + 128, invn);
        // BN2 (+relu except last layer) -> h (or d_out node region), pool into fp
        float* hout = (l == Ll - 1) ? out : h;
        bn_pool_kernel<<<(ND + 255) / 256, 256, 0, stream>>>(
            z, scsh + 128, batch, hout, fp, l, (l != Ll - 1) ? 1 : 0);
    }
}